// Model_19155554140252
// MI455X (gfx1250) — compile-verified
//
#include <hip/hip_runtime.h>
#include <hip/hip_bf16.h>

typedef __attribute__((ext_vector_type(2))) float v2f;
typedef __attribute__((ext_vector_type(8))) float v8f;
typedef __attribute__((ext_vector_type(4))) int v4i;

#define N_NODES 100000
#define EPS 1e-5f

// -------- CDNA5 async global->LDS copy (guarded; falls back to sync copy) ---
#if defined(__gfx1250__) && __has_builtin(__builtin_amdgcn_global_load_async_to_lds_b128)
#define ASYNC_COPY 1
// Address-space-qualified pointer types (clang prints AS1 as __device__,
// AS3 as __shared__ in HIP mode).
typedef __attribute__((address_space(1))) v4i gv4i;
typedef __attribute__((address_space(3))) v4i lv4i;
#else
#define ASYNC_COPY 0
#endif

template <int PENDING>
__device__ __forceinline__ void wait_async_le() {
#if defined(__gfx1250__)
#if __has_builtin(__builtin_amdgcn_s_wait_asynccnt)
  __builtin_amdgcn_s_wait_asynccnt((short)PENDING);
#else
  asm volatile("s_wait_asynccnt %0" ::"i"(PENDING) : "memory");
#endif
#endif
}

// ---------------------------------------------------------------------------
// GEMM: C[n_rows, NCOLS] = A[n_rows, K] @ B[K, NCOLS], all fp32, using
// V_WMMA_F32_16X16X4_F32. Each workgroup processes TM consecutive 16-row
// M-tiles with a double-buffered LDS A-tile filled by async global->LDS
// B128 loads (ASYNCcnt pipeline), so tile t+1 streams in while the WMMAs
// consume tile t. Each wave owns NCOLS/16/WAVES 16x16 output tiles.
//
// f32 WMMA operand layout (ISA 7.12.2):
//   A (16x4):  lane L (L%16 = M row). VGPR0 = K=(L/16)*2, VGPR1 = +1
//   B (4x16):  lane L (L%16 = N col). VGPR0 = K=(L/16)*2, VGPR1 = +1
//   C/D (16x16): VGPR r, lanes 0-15 -> M=r, lanes 16-31 -> M=r+8, N = L%16
// ---------------------------------------------------------------------------
template <int K, int NCOLS, int WAVES, int TM>
__global__ void gemm_wmma_f32(const float* __restrict__ A,
                              const float* __restrict__ B,
                              float* __restrict__ C) {
  constexpr int NTHREADS = WAVES * 32;
  constexpr int TILE_ELEMS = 16 * K;
  constexpr int LOADS = TILE_ELEMS / (NTHREADS * 4);  // b128 loads per thread

  __shared__ float lds_a[2][TILE_ELEMS];
  const int tid = threadIdx.x;
  const int tile0 = blockIdx.x * TM;

  // Issue the (async) copy of m-tile (tile0 + t) into LDS buffer `buf`.
  auto issue = [&](int t, int buf) {
    const float* src = A + (size_t)(tile0 + t) * TILE_ELEMS;
#pragma unroll
    for (int i = 0; i < LOADS; ++i) {
      const int idx = (tid + i * NTHREADS) * 4;
#if ASYNC_COPY
      __builtin_amdgcn_global_load_async_to_lds_b128(
          (gv4i*)(src + idx), (lv4i*)(&lds_a[buf][idx]), 0, 0);
#else
      const float4 v = *reinterpret_cast<const float4*>(src + idx);
      *reinterpret_cast<float4*>(&lds_a[buf][idx]) = v;
#endif
    }
  };

  const int wave = tid >> 5;
  const int lane = tid & 31;
  const int lm   = lane & 15;        // M row (A) / N col (B,C) within tile
  const int koff = (lane >> 4) * 2;  // K pair selector
  constexpr int NT = NCOLS / 16;
  constexpr int TILES_PER_WAVE = NT / WAVES;

  issue(0, 0);

  for (int t = 0; t < TM; ++t) {
    const int buf = t & 1;
    if (t + 1 < TM) {
      issue(t + 1, buf ^ 1);
      wait_async_le<LOADS>();  // first LOADS (tile t) landed; t+1 in flight
    } else {
      wait_async_le<0>();
    }
    __syncthreads();

    const int m0 = (tile0 + t) * 16;
#pragma unroll
    for (int nt_i = 0; nt_i < TILES_PER_WAVE; ++nt_i) {
      const int n0 = (wave * TILES_PER_WAVE + nt_i) * 16;
      v8f c = {};
      for (int k = 0; k < K; k += 4) {
        v2f a, b;
        a.x = lds_a[buf][lm * K + k + koff];
        a.y = lds_a[buf][lm * K + k + koff + 1];
        b.x = B[(size_t)(k + koff) * NCOLS + n0 + lm];
        b.y = B[(size_t)(k + koff + 1) * NCOLS + n0 + lm];
        c = __builtin_amdgcn_wmma_f32_16x16x4_f32(
            /*neg_a=*/false, a, /*neg_b=*/false, b,
            /*c_mod=*/(short)0, c, /*reuse_a=*/false, /*reuse_b=*/false);
      }
      const int mbase = m0 + (lane >> 4) * 8;
#pragma unroll
      for (int r = 0; r < 8; ++r) {
        C[(size_t)(mbase + r) * NCOLS + n0 + lm] = c[r];
      }
    }
    __syncthreads();  // everyone done with lds_a[buf] before it is refilled
  }
}

// ---------------------------------------------------------------------------
// SpMM: out[row[e]] += val[e] * dense[col[e]].
// One wave handles 32 edges per batch: edge metadata is loaded coalesced
// (one edge per lane) and broadcast with __shfl; the feature row is gathered
// with float4/float2 vector loads (dense is L2-resident: <=102 MB vs 192 MB
// L2) and accumulated with fp32 global atomics resolved in L2.
// ---------------------------------------------------------------------------
template <int F>
__global__ void spmm_atomic(const int* __restrict__ rows,
                            const int* __restrict__ cols,
                            const float* __restrict__ vals,
                            const float* __restrict__ dense,
                            float* __restrict__ out, int E) {
  const int lane   = threadIdx.x & 31;
  const int gwave  = (blockIdx.x * blockDim.x + threadIdx.x) >> 5;
  const int nwaves = (gridDim.x * blockDim.x) >> 5;

  for (int base = gwave * 32; base < E; base += nwaves * 32) {
    const int e = base + lane;
    int r_l = 0, c_l = 0;
    float v_l = 0.f;
    if (e < E) {
      r_l = rows[e];
      c_l = cols[e];
      v_l = vals[e];
    }
    const int cnt = (E - base < 32) ? (E - base) : 32;
    for (int i = 0; i < cnt; ++i) {
      const int r   = __shfl(r_l, i);
      const int c   = __shfl(c_l, i);
      const float v = __shfl(v_l, i);
      const float* src = dense + (size_t)c * F;
      float* dst       = out   + (size_t)r * F;
      if (F == 256) {
#pragma unroll
        for (int half = 0; half < 2; ++half) {
          const int j = (lane + half * 32) * 4;
          const float4 s = *reinterpret_cast<const float4*>(src + j);
          unsafeAtomicAdd(&dst[j + 0], v * s.x);
          unsafeAtomicAdd(&dst[j + 1], v * s.y);
          unsafeAtomicAdd(&dst[j + 2], v * s.z);
          unsafeAtomicAdd(&dst[j + 3], v * s.w);
        }
      } else {  // F == 64
        const int j = lane * 2;
        const float2 s = *reinterpret_cast<const float2*>(src + j);
        unsafeAtomicAdd(&dst[j + 0], v * s.x);
        unsafeAtomicAdd(&dst[j + 1], v * s.y);
      }
    }
  }
}

// ---------------------------------------------------------------------------
// BatchNorm pass 1: per-column sum / sum-of-squares of relu(h).
// ---------------------------------------------------------------------------
__global__ void bn_stats_relu(const float* __restrict__ h, int n_rows, int F,
                              float* __restrict__ sums,
                              float* __restrict__ sumsq) {
  const int c = threadIdx.x;
  const int chunk = (n_rows + gridDim.x - 1) / gridDim.x;
  const int r0 = blockIdx.x * chunk;
  const int r1 = (r0 + chunk < n_rows) ? (r0 + chunk) : n_rows;
  float s = 0.f, s2 = 0.f;
  for (int r = r0; r < r1; ++r) {
    float v = h[(size_t)r * F + c];
    v = fmaxf(v, 0.f);  // relu precedes batchnorm in the reference
    s += v;
    s2 += v * v;
  }
  unsafeAtomicAdd(&sums[c], s);
  unsafeAtomicAdd(&sumsq[c], s2);
}

// BatchNorm pass 2a: fold (mean, var, gamma, beta) into per-column scale/shift.
__global__ void bn_finalize(const float* __restrict__ sums,
                            const float* __restrict__ sumsq,
                            const float* __restrict__ gamma,
                            const float* __restrict__ beta,
                            float* __restrict__ scale,
                            float* __restrict__ shift, int F, float inv_n) {
  const int c = threadIdx.x;
  if (c < F) {
    const float mean = sums[c] * inv_n;
    const float var  = sumsq[c] * inv_n - mean * mean;
    const float sc   = gamma[c] * rsqrtf(var + EPS);
    scale[c] = sc;
    shift[c] = beta[c] - mean * sc;
  }
}

// BatchNorm pass 2b: y = [relu?]( relu(x) * scale[c] + shift[c] )
__global__ void bn_apply(const float* __restrict__ in, float* __restrict__ out,
                         const float* __restrict__ scale,
                         const float* __restrict__ shift,
                         size_t total, int Fmask, int relu_out) {
  size_t i = (size_t)blockIdx.x * blockDim.x + threadIdx.x;
  const size_t stride = (size_t)gridDim.x * blockDim.x;
  for (; i < total; i += stride) {
    const int c = (int)(i & (size_t)Fmask);  // F is a power of two
    float v = fmaxf(in[i], 0.f);
    v = v * scale[c] + shift[c];
    if (relu_out) v = fmaxf(v, 0.f);
    out[i] = v;
  }
}

// ---------------------------------------------------------------------------
// Orchestration. Workspace layout (floats):
//   [0, N*256)            H0  : X@W1, later reused as H1N (post-BN layer1 act)
//   [N*256, 2N*256)       AGG1: spmm layer-1 accumulator (zeroed)
//   [2N*256, 2N*256+N*64) H2  : H1N@W2
//   tail                  stats: sums1/sumsq1/scale1/shift1 (256 ea),
//                                sums2/sumsq2/scale2/shift2 (64 ea)
// ---------------------------------------------------------------------------
extern "C" void kernel_launch(void* const* d_in, const int* in_sizes, int n_in,
                              void* d_out, int out_size, void* d_ws,
                              size_t ws_size, hipStream_t stream) {
  const float* x    = (const float*)d_in[0];
  const int*   erow = (const int*)d_in[1];
  const int*   ecol = (const int*)d_in[2];
  const float* eval = (const float*)d_in[3];
  const float* W1   = (const float*)d_in[4];
  const float* g1   = (const float*)d_in[5];
  const float* b1   = (const float*)d_in[6];
  const float* W2   = (const float*)d_in[7];
  const float* g2   = (const float*)d_in[8];
  const float* b2   = (const float*)d_in[9];
  float* out = (float*)d_out;

  const int N = N_NODES;
  const int E = in_sizes[1];

  float* H0   = (float*)d_ws;
  float* AGG1 = H0 + (size_t)N * 256;
  float* H2   = AGG1 + (size_t)N * 256;
  float* st   = H2 + (size_t)N * 64;
  float* sums1 = st;         float* sumsq1 = st + 256;
  float* scale1 = st + 512;  float* shift1 = st + 768;
  float* sums2 = st + 1024;  float* sumsq2 = st + 1088;
  float* scale2 = st + 1152; float* shift2 = st + 1216;

  // Zero the accumulators (graph-capture-safe async memsets).
  (void)hipMemsetAsync(AGG1, 0, (size_t)N * 256 * sizeof(float), stream);
  (void)hipMemsetAsync(out, 0, (size_t)N * 64 * sizeof(float), stream);
  (void)hipMemsetAsync(st, 0, 1280 * sizeof(float), stream);

  // ---- Layer 1 ----  (6250 M-tiles; 2 per block, double-buffered)
  gemm_wmma_f32<512, 256, 8, 2><<<N / 32, 256, 0, stream>>>(x, W1, H0);
  spmm_atomic<256><<<2048, 256, 0, stream>>>(erow, ecol, eval, H0, AGG1, E);
  bn_stats_relu<<<256, 256, 0, stream>>>(AGG1, N, 256, sums1, sumsq1);
  bn_finalize<<<1, 256, 0, stream>>>(sums1, sumsq1, g1, b1, scale1, shift1, 256,
                                     1.0f / (float)N);
  bn_apply<<<2048, 256, 0, stream>>>(AGG1, H0, scale1, shift1, (size_t)N * 256,
                                     255, 1);

  // ---- Layer 2 ----
  gemm_wmma_f32<256, 64, 4, 2><<<N / 32, 128, 0, stream>>>(H0, W2, H2);
  spmm_atomic<64><<<2048, 256, 0, stream>>>(erow, ecol, eval, H2, out, E);
  bn_stats_relu<<<256, 64, 0, stream>>>(out, N, 64, sums2, sumsq2);
  bn_finalize<<<1, 64, 0, stream>>>(sums2, sumsq2, g2, b2, scale2, shift2, 64,
                                    1.0f / (float)N);
  bn_apply<<<2048, 256, 0, stream>>>(out, out, scale2, shift2, (size_t)N * 64,
                                     63, 0);
}